// DecodePredictions_72344429134389
// MI455X (gfx1250) — compile-verified
//
#include <hip/hip_runtime.h>
#include <hip/hip_bf16.h>
#include <stdint.h>

#define NUM_ANCHORS   76725
#define NUM_CLASSES   80
#define BATCH_N       8
#define PRED_STRIDE   84      // 4 box + 80 class logits
#define PRE_NMS_K     200
#define MAX_PER_CLASS 100
#define MAX_DET       100
#define CONF_THR      0.05f
#define IOU_THR       0.5f
#define GATHER_CAP    512     // >= PRE_NMS_K + histogram-boundary slack
#define ANCH_TILE     1024    // anchors per decode workgroup (one TDM descriptor)

#define HAVE_TDM 0
#if defined(__has_builtin)
#if __has_builtin(__builtin_amdgcn_tensor_load_to_lds)
#undef HAVE_TDM
#define HAVE_TDM 1
#endif
#endif

typedef unsigned int v4u_t __attribute__((ext_vector_type(4)));
typedef int          v8i_t __attribute__((ext_vector_type(8)));
typedef int          v4i_t __attribute__((ext_vector_type(4)));

// ---------------------------------------------------------------------------
// Kernel: decode corner boxes. The box-regression data is a strided 2D tile
// (n rows x 4 floats, row stride 84 floats) -> exactly the TDM use case.
// Wave 0 issues ONE tensor_load_to_lds descriptor covering 1024 anchors,
// waits TENSORcnt, barrier, then 256 lanes decode 4 anchors each from LDS.
// Fallback (no TDM builtin): per-lane global_load_async_to_lds_b128.
// ---------------------------------------------------------------------------
__global__ __launch_bounds__(256) void decode_kernel(
    const float* __restrict__ pred, const float* __restrict__ anchors,
    float* __restrict__ boxes) {
    __shared__ float sbox[ANCH_TILE * 4];              // 16 KB
    const int tid = threadIdx.x;
    const int a0  = blockIdx.x * ANCH_TILE;
    const int b   = blockIdx.y;
    const int rem = NUM_ANCHORS - a0;
    const unsigned n = (unsigned)(rem < ANCH_TILE ? rem : ANCH_TILE);

#if HAVE_TDM
    if (tid < 32) {   // wave 0 issues the tile DMA for the whole workgroup
        const uint64_t gaddr =
            (uint64_t)(uintptr_t)(pred + ((size_t)b * NUM_ANCHORS + (size_t)a0) * PRED_STRIDE);
        const uint32_t lds0 = (uint32_t)(uintptr_t)&sbox[0];
        // D# group 0: count=1, lds_addr, global_addr[56:0], type=2
        v4u_t g0;
        g0[0] = 1u;
        g0[1] = lds0;
        g0[2] = (unsigned)(gaddr & 0xffffffffu);
        g0[3] = (unsigned)((gaddr >> 32) & 0x1ffffffu) | 0x80000000u;
        // D# group 1: data_size=4B, tensor_dim0=84, tensor_dim1=n,
        //             tile_dim0=4, tile_dim1=n, tensor_dim0_stride=84
        v8i_t g1;
        g1[0] = (int)(2u << 16);            // data_size code 2 (4 bytes)
        g1[1] = (int)(84u << 16);           // tensor_dim0[15:0] @ bits 63:48
        g1[2] = (int)(n << 16);             // tensor_dim1[15:0] @ bits 95:80
        g1[3] = (int)(4u << 16);            // tile_dim0 @ bits 127:112
        g1[4] = (int)n;                     // tile_dim1 @ bits 143:128 (tile_dim2=0)
        g1[5] = 84;                         // tensor_dim0_stride[31:0]
        g1[6] = 0;
        g1[7] = 0;
        v4i_t gz = {0, 0, 0, 0};
#if __clang_major__ >= 23
        v8i_t gz8 = {0, 0, 0, 0, 0, 0, 0, 0};
        __builtin_amdgcn_tensor_load_to_lds(g0, g1, gz, gz, gz8, 0);
#else
        __builtin_amdgcn_tensor_load_to_lds(g0, g1, gz, gz, 0);
#endif
        __builtin_amdgcn_s_wait_tensorcnt(0);
    }
    __syncthreads();
#else
#pragma unroll
    for (int u = 0; u < 4; ++u) {
        const int a = a0 + u * 256 + tid;
        if (a < NUM_ANCHORS) {
            const float* src = pred + ((size_t)b * NUM_ANCHORS + (size_t)a) * PRED_STRIDE;
            uint32_t laddr = (uint32_t)(uintptr_t)(&sbox[(u * 256 + tid) * 4]);
            uint64_t gaddr = (uint64_t)(uintptr_t)src;
            asm volatile("global_load_async_to_lds_b128 %0, %1, off"
                         :: "v"(laddr), "v"(gaddr) : "memory");
        }
    }
    asm volatile("s_wait_asynccnt 0x0" ::: "memory");
#endif

#pragma unroll
    for (int u = 0; u < 4; ++u) {
        const int r = u * 256 + tid;
        const int a = a0 + r;
        if (a >= NUM_ANCHORS) break;
        volatile float* vb = &sbox[r * 4];
        const float t0 = vb[0] * 0.1f, t1 = vb[1] * 0.1f;
        const float t2 = vb[2] * 0.2f, t3 = vb[3] * 0.2f;
        const float4 an = *reinterpret_cast<const float4*>(anchors + (size_t)a * 4);
        const float cx = t0 * an.z + an.x;
        const float cy = t1 * an.w + an.y;
        const float w  = expf(t2) * an.z;
        const float h  = expf(t3) * an.w;
        float4 ob;
        ob.x = cx - 0.5f * w; ob.y = cy - 0.5f * h;
        ob.z = cx + 0.5f * w; ob.w = cy + 0.5f * h;
        *reinterpret_cast<float4*>(boxes + ((size_t)b * NUM_ANCHORS + a) * 4) = ob;
    }
}

// ---------------------------------------------------------------------------
// Async column streamer: double-buffered global_load_async_to_lds_b32.
// Each wave keeps <=8 async loads in flight (2 tiles x 4), blocking only on
// s_wait_asynccnt 0x4. Tail lanes clamp the address (load always issues, so
// ASYNCcnt bookkeeping is exact) and are masked out of processing only.
// Lanes read back only their own slots -> no cross-wave barriers needed.
// ---------------------------------------------------------------------------
template <typename F>
__device__ __forceinline__ void stream_column(const float* __restrict__ colbase,
                                              float* __restrict__ sstage,
                                              int tid, F fn) {
    constexpr int TILE = 1024;                       // 256 lanes * 4 elements
    constexpr int NT   = (NUM_ANCHORS + TILE - 1) / TILE;
    auto issue = [&](int buf, int t) {
#pragma unroll
        for (int u = 0; u < 4; ++u) {
            int a = t * TILE + u * 256 + tid;
            a = (a < NUM_ANCHORS) ? a : (NUM_ANCHORS - 1);
            const float* src = colbase + (size_t)a * PRED_STRIDE;
            uint32_t laddr = (uint32_t)(uintptr_t)&sstage[buf * TILE + u * 256 + tid];
            asm volatile("global_load_async_to_lds_b32 %0, %1, off"
                         :: "v"(laddr), "v"((uint64_t)(uintptr_t)src) : "memory");
        }
    };
    issue(0, 0);
    for (int t = 0; t < NT; ++t) {
        const int buf = t & 1;
        if (t + 1 < NT) {
            issue(buf ^ 1, t + 1);
            asm volatile("s_wait_asynccnt 0x4" ::: "memory");  // previous tile done
        } else {
            asm volatile("s_wait_asynccnt 0x0" ::: "memory");
        }
#pragma unroll
        for (int u = 0; u < 4; ++u) {
            const int a = t * TILE + u * 256 + tid;
            if (a < NUM_ANCHORS) fn(a, sstage[buf * TILE + u * 256 + tid]);
        }
    }
}

// ---------------------------------------------------------------------------
// Kernel: per (image,class): exact top-200 on logits (sigmoid is monotone).
// One 4096-bin LDS histogram locates the top-200 boundary; a refine pass over
// 256 sub-bins runs only if the boundary bin would overflow the gather buffer.
// Then gather + bitonic sort 512 + greedy NMS + compact to first 100 kept.
// ---------------------------------------------------------------------------
__global__ __launch_bounds__(256) void topk_nms_kernel(
    const float* __restrict__ pred, const float* __restrict__ boxes,
    float* __restrict__ cls_boxes, float* __restrict__ cls_scores) {
    const int c = blockIdx.x, b = blockIdx.y, tid = threadIdx.x;

    __shared__ float sstage[2 * 1024];               // async staging (8 KB)
    __shared__ int   hist4k[4096];                   // 16 KB
    __shared__ int   part[256];                      // chunk sums / sub-hist
    __shared__ float gsc[GATHER_CAP];
    __shared__ int   gidx[GATHER_CAP];
    __shared__ float sbx[PRE_NMS_K * 4];
    __shared__ int   skeep[PRE_NMS_K];
    __shared__ int   sorder[MAX_PER_CLASS];
    __shared__ int   sB, sAbove, sJ2, sNeedRef, scnt, snkept;

    const float LO    = -14.0f;
    const float INVW1 = 4096.0f / 28.0f;             // 1 / bin width (4096 bins)
    const float W1    = 28.0f / 4096.0f;
    const float INVW2 = 1048576.0f / 28.0f;          // 1 / sub-bin width (x256)
    const float* colbase =
        pred + ((size_t)b * NUM_ANCHORS) * PRED_STRIDE + 4 + (size_t)c;

    // ---- pass 1: 4096-bin histogram over logits ----
    for (int i = tid; i < 4096; i += 256) hist4k[i] = 0;
    __syncthreads();
    stream_column(colbase, sstage, tid, [&](int /*a*/, float v) {
        int bin = (int)((v - LO) * INVW1);
        bin = bin < 0 ? 0 : (bin > 4095 ? 4095 : bin);
        atomicAdd(&hist4k[bin], 1);
    });
    __syncthreads();
    // parallel suffix locate: 256 chunk sums, short serial scans
    {
        int ps = 0;
#pragma unroll
        for (int i = 0; i < 16; ++i) ps += hist4k[tid * 16 + i];
        part[tid] = ps;
    }
    __syncthreads();
    if (tid == 0) {
        int cum = 0, Bc = 0;
        for (int i = 255; i >= 0; --i) { cum += part[i]; if (cum >= PRE_NMS_K) { Bc = i; break; } }
        int above = cum - part[Bc];                 // strictly above chunk Bc
        int B = Bc * 16;
        for (int i = Bc * 16 + 15; i >= Bc * 16; --i) {
            above += hist4k[i];
            if (above >= PRE_NMS_K) { B = i; break; }
        }
        const int occB = hist4k[B];
        sB = B; sAbove = above - occB;              // strictly above bin B
        sNeedRef = ((above - occB) + occB > GATHER_CAP) ? 1 : 0;
        sJ2 = 0; scnt = 0;
    }
    __syncthreads();
    const int  B       = sB;
    const int  above   = sAbove;
    const bool refined = (sNeedRef != 0);           // WG-uniform
    const float lo1 = LO + (float)B * W1;

    // ---- optional refine pass: 256 sub-bins inside boundary bin ----
    if (refined) {
        part[tid] = 0;
        __syncthreads();
        stream_column(colbase, sstage, tid, [&](int /*a*/, float v) {
            int bin = (int)((v - LO) * INVW1);
            bin = bin < 0 ? 0 : (bin > 4095 ? 4095 : bin);
            if (bin == B) {
                int sub = (int)((v - lo1) * INVW2);
                sub = sub < 0 ? 0 : (sub > 255 ? 255 : sub);
                atomicAdd(&part[sub], 1);
            }
        });
        __syncthreads();
        if (tid == 0) {
            int cum = above, j2 = 0;
            for (int i = 255; i >= 0; --i) { cum += part[i]; if (cum >= PRE_NMS_K) { j2 = i; break; } }
            sJ2 = j2;
        }
        __syncthreads();
    }
    const int j2 = sJ2;

    // ---- gather everything at/above the threshold ----
    stream_column(colbase, sstage, tid, [&](int a, float v) {
        int bin = (int)((v - LO) * INVW1);
        bin = bin < 0 ? 0 : (bin > 4095 ? 4095 : bin);
        bool take;
        if (refined) {
            take = bin > B;
            if (bin == B) {
                int sub = (int)((v - lo1) * INVW2);
                sub = sub < 0 ? 0 : (sub > 255 ? 255 : sub);
                take = (sub >= j2);
            }
        } else {
            take = (bin >= B);
        }
        if (take) {
            const int p = atomicAdd(&scnt, 1);
            if (p < GATHER_CAP) { gsc[p] = v; gidx[p] = a; }
        }
    });
    __syncthreads();
    const int got = scnt < GATHER_CAP ? scnt : GATHER_CAP;
    for (int i = tid; i < GATHER_CAP; i += 256)
        if (i >= got) { gsc[i] = -1.0e30f; gidx[i] = 0; }
    __syncthreads();

    // ---- bitonic sort 512 (descending by logit, index payload) ----
    for (int k = 2; k <= GATHER_CAP; k <<= 1) {
        for (int j = k >> 1; j > 0; j >>= 1) {
            __syncthreads();
            for (int bb = 0; bb < GATHER_CAP; bb += 256) {
                const int i = bb + tid;
                const int ixj = i ^ j;
                if (ixj > i) {
                    const bool desc = ((i & k) == 0);
                    const float va = gsc[i], vbv = gsc[ixj];
                    if ((va < vbv) == desc) {
                        gsc[i] = vbv; gsc[ixj] = va;
                        const int t = gidx[i]; gidx[i] = gidx[ixj]; gidx[ixj] = t;
                    }
                }
            }
        }
    }
    __syncthreads();

    // ---- load candidate boxes (b128), init keep mask with conf threshold ----
    for (int r = tid; r < PRE_NMS_K; r += 256) {
        const int ai = gidx[r];
        const float4 bb4 =
            *reinterpret_cast<const float4*>(boxes + ((size_t)b * NUM_ANCHORS + ai) * 4);
        sbx[r * 4 + 0] = bb4.x; sbx[r * 4 + 1] = bb4.y;
        sbx[r * 4 + 2] = bb4.z; sbx[r * 4 + 3] = bb4.w;
        const float sc = 1.0f / (1.0f + expf(-gsc[r]));
        skeep[r] = (sc > CONF_THR) ? 1 : 0;
    }

    // ---- greedy NMS (matches reference fori_loop suppression) ----
    for (int i = 0; i < PRE_NMS_K; ++i) {
        __syncthreads();
        if (!skeep[i]) continue;                  // uniform shared read
        const int j = tid;
        if (j > i && j < PRE_NMS_K && skeep[j]) {
            const float x1 = fmaxf(sbx[i * 4 + 0], sbx[j * 4 + 0]);
            const float y1 = fmaxf(sbx[i * 4 + 1], sbx[j * 4 + 1]);
            const float x2 = fminf(sbx[i * 4 + 2], sbx[j * 4 + 2]);
            const float y2 = fminf(sbx[i * 4 + 3], sbx[j * 4 + 3]);
            const float iw = fmaxf(x2 - x1, 0.0f), ih = fmaxf(y2 - y1, 0.0f);
            const float inter = iw * ih;
            const float ai_ = (sbx[i * 4 + 2] - sbx[i * 4 + 0]) * (sbx[i * 4 + 3] - sbx[i * 4 + 1]);
            const float aj_ = (sbx[j * 4 + 2] - sbx[j * 4 + 0]) * (sbx[j * 4 + 3] - sbx[j * 4 + 1]);
            const float iou = inter / (ai_ + aj_ - inter + 1e-8f);
            if (iou > IOU_THR) skeep[j] = 0;
        }
    }
    __syncthreads();

    // ---- compact first 100 kept (desc order preserved) ----
    if (tid == 0) {
        int w = 0;
        for (int r = 0; r < PRE_NMS_K && w < MAX_PER_CLASS; ++r)
            if (skeep[r]) sorder[w++] = r;
        snkept = w;
    }
    __syncthreads();
    float* ob = cls_boxes  + ((size_t)(b * NUM_CLASSES + c)) * MAX_PER_CLASS * 4;
    float* os = cls_scores + ((size_t)(b * NUM_CLASSES + c)) * MAX_PER_CLASS;
    for (int k = tid; k < MAX_PER_CLASS; k += 256) {
        float4 v4 = {0.f, 0.f, 0.f, 0.f};
        float  sv = 0.f;
        if (k < snkept) {
            const int r = sorder[k];
            sv = 1.0f / (1.0f + expf(-gsc[r]));
            v4.x = sbx[r * 4 + 0]; v4.y = sbx[r * 4 + 1];
            v4.z = sbx[r * 4 + 2]; v4.w = sbx[r * 4 + 3];
        }
        os[k] = sv;
        *reinterpret_cast<float4*>(ob + k * 4) = v4;
    }
}

// ---------------------------------------------------------------------------
// Kernel: regenerate anchors [A,4] = (cx, cy, w, h)
// ---------------------------------------------------------------------------
__global__ __launch_bounds__(256) void anchors_kernel(float* __restrict__ anchors) {
    const int a = blockIdx.x * 256 + threadIdx.x;
    if (a >= NUM_ANCHORS) return;
    int lvl, base;
    if      (a < 57600) { lvl = 0; base = 0; }
    else if (a < 72000) { lvl = 1; base = 57600; }
    else if (a < 75600) { lvl = 2; base = 72000; }
    else if (a < 76500) { lvl = 3; base = 75600; }
    else                { lvl = 4; base = 76500; }
    const int idx    = a - base;
    const int fw     = 80 >> lvl;
    const int stride = 8 << lvl;
    const int cell = idx / 9, k = idx % 9;
    const int y = cell / fw, x = cell % fw;
    const float cx = (x + 0.5f) * (float)stride;
    const float cy = (y + 0.5f) * (float)stride;
    const float side = (float)(32 << lvl);                    // sqrt(area)
    const int ri = k / 3, si = k % 3;
    const float sr = (ri == 0) ? 0.70710678118f : ((ri == 1) ? 1.0f : 1.41421356237f);
    const float s  = exp2f((float)si * (1.0f / 3.0f));
    float4 o;
    o.x = cx; o.y = cy;
    o.z = s * (side * sr);       // w = area / h
    o.w = s * (side / sr);       // h = sqrt(area / r)
    *reinterpret_cast<float4*>(anchors + (size_t)a * 4) = o;
}

// ---------------------------------------------------------------------------
// Kernel: per image, iterative argmax top-100 over 8000 candidates.
// ---------------------------------------------------------------------------
__global__ __launch_bounds__(256) void merge_kernel(
    const float* __restrict__ cls_boxes, const float* __restrict__ cls_scores,
    float* __restrict__ out_boxes, float* __restrict__ out_scores,
    float* __restrict__ out_classes, float* __restrict__ out_valid) {
    const int b = blockIdx.x, tid = threadIdx.x;
    const int NCAND = NUM_CLASSES * MAX_PER_CLASS;   // 8000
    __shared__ float ss[8192];
    __shared__ float rmax[256];
    __shared__ int   ridx[256];
    const float* cs = cls_scores + (size_t)b * NCAND;
    const float* cb = cls_boxes  + (size_t)b * NCAND * 4;
    for (int i = tid; i < 8192; i += 256) ss[i] = (i < NCAND) ? cs[i] : -1.0f;
    __syncthreads();

    int cnt = 0;   // thread 0 only
    for (int sel = 0; sel < MAX_DET; ++sel) {
        float m = -2.0f; int mi = 0;
        for (int i = tid; i < 8192; i += 256) {
            const float v = ss[i];
            if (v > m) { m = v; mi = i; }
        }
        rmax[tid] = m; ridx[tid] = mi;
        __syncthreads();
        for (int s = 128; s > 0; s >>= 1) {
            if (tid < s && rmax[tid + s] > rmax[tid]) {
                rmax[tid] = rmax[tid + s]; ridx[tid] = ridx[tid + s];
            }
            __syncthreads();
        }
        if (tid == 0) {
            const float sc = rmax[0]; const int id = ridx[0];
            const bool pos = sc > 0.0f;
            out_scores [b * MAX_DET + sel] = pos ? sc : 0.0f;
            out_classes[b * MAX_DET + sel] = pos ? (float)(id / MAX_PER_CLASS) : 0.0f;
            float4 v4 = {0.f, 0.f, 0.f, 0.f};
            if (pos) {
                v4 = *reinterpret_cast<const float4*>(cb + (size_t)id * 4);
                ++cnt;
            }
            *reinterpret_cast<float4*>(out_boxes + ((size_t)b * MAX_DET + sel) * 4) = v4;
            ss[id] = -2.0f;
        }
        __syncthreads();
    }
    if (tid == 0) out_valid[b] = (float)cnt;
}

// ---------------------------------------------------------------------------
extern "C" void kernel_launch(void* const* d_in, const int* in_sizes, int n_in,
                              void* d_out, int out_size, void* d_ws, size_t ws_size,
                              hipStream_t stream) {
    const float* pred = (const float*)d_in[1];   // d_in[0] = images (unused, static shape)

    const size_t anchors_n   = (size_t)NUM_ANCHORS * 4;
    const size_t boxes_n     = (size_t)BATCH_N * NUM_ANCHORS * 4;
    const size_t cls_boxes_n = (size_t)BATCH_N * NUM_CLASSES * MAX_PER_CLASS * 4;
    const size_t cls_score_n = (size_t)BATCH_N * NUM_CLASSES * MAX_PER_CLASS;
    const size_t need = (anchors_n + boxes_n + cls_boxes_n + cls_score_n) * sizeof(float);
    if (ws_size < need) return;

    float* ws         = (float*)d_ws;
    float* anchors    = ws;
    float* boxes      = anchors + anchors_n;
    float* cls_boxes  = boxes + boxes_n;
    float* cls_scores = cls_boxes + cls_boxes_n;

    float* out         = (float*)d_out;
    float* out_boxes   = out;                                   // [8,100,4]
    float* out_scores  = out_boxes + BATCH_N * MAX_DET * 4;     // [8,100]
    float* out_classes = out_scores + BATCH_N * MAX_DET;        // [8,100]
    float* out_valid   = out_classes + BATCH_N * MAX_DET;       // [8]

    const int tiles256  = (NUM_ANCHORS + 255) / 256;
    const int tiles1024 = (NUM_ANCHORS + ANCH_TILE - 1) / ANCH_TILE;
    anchors_kernel<<<tiles256, 256, 0, stream>>>(anchors);
    decode_kernel<<<dim3(tiles1024, BATCH_N), 256, 0, stream>>>(pred, anchors, boxes);
    topk_nms_kernel<<<dim3(NUM_CLASSES, BATCH_N), 256, 0, stream>>>(pred, boxes, cls_boxes, cls_scores);
    merge_kernel<<<BATCH_N, 256, 0, stream>>>(cls_boxes, cls_scores,
                                              out_boxes, out_scores, out_classes, out_valid);
}